// Attention_K_Layer_10960756539882
// MI455X (gfx1250) — compile-verified
//
#include <hip/hip_runtime.h>
#include <math.h>

typedef __attribute__((ext_vector_type(2))) float v2f;
typedef __attribute__((ext_vector_type(8))) float v8f;

#define B_  64
#define T_  512
#define F_  768
#define NG  8          // 8 used gate columns: 0..3 fwd(i,f,g,o), 4..7 bwd
#define THRESH_ 0.1f

// ---------------------------------------------------------------------------
// Kernel 1: gate pre-activations via V_WMMA_F32_16X16X4_F32.
// rows = B*T = 32768 flattened (b,t). Each wave computes a 16x16 tile:
// 16 (b,t) rows x 16 gate columns (8 valid, rest multiply by zero weights).
// A fragment (16x4 f32): lanes 0-15 hold K={0,1}, lanes 16-31 hold K={2,3},
// M = lane&15 -> one contiguous float2 load from x per lane per step.
// B fragment (4x16 f32): lanes 0-15 VGPR0/1 = K0/K1, lanes 16-31 = K2/K3,
// N = lane&15 -> two LDS reads from the packed weight tile.
// ---------------------------------------------------------------------------
__global__ __launch_bounds__(256)
void gates_gemm_wmma(const float* __restrict__ x,
                     const float* __restrict__ Wih_f,   // [4, F]
                     const float* __restrict__ Wih_b,   // [4, F]
                     float* __restrict__ gates)         // [B*T, NG]
{
    __shared__ float Wp[F_ * 16];   // 48 KB: Wp[k][n], n<4 fwd, n<8 bwd, else 0
    for (int i = threadIdx.x; i < F_ * 16; i += 256) {
        const int k = i >> 4, n = i & 15;
        float v = 0.0f;
        if (n < 4)      v = Wih_f[n * F_ + k];
        else if (n < 8) v = Wih_b[(n - 4) * F_ + k];
        Wp[i] = v;
    }
    __syncthreads();

    const int wave    = threadIdx.x >> 5;
    const int lane    = threadIdx.x & 31;
    const int laneHi  = lane >> 4;          // 0: K pair {0,1}; 1: K pair {2,3}
    const int n       = lane & 15;          // A: row M; B/C/D: col N
    const int rowBase = blockIdx.x * 128 + wave * 16;

    const float* xrow = x + (size_t)(rowBase + n) * F_ + 2 * laneHi;

    v8f c = {};
    #pragma unroll 4
    for (int k0 = 0; k0 < F_; k0 += 4) {
        const v2f a = *(const v2f*)(xrow + k0);     // 8B aligned, coalesced
        const int kk = k0 + 2 * laneHi;
        v2f b;
        b.x = Wp[kk * 16 + n];
        b.y = Wp[kk * 16 + 16 + n];
        c = __builtin_amdgcn_wmma_f32_16x16x4_f32(
                /*neg_a=*/false, a, /*neg_b=*/false, b,
                /*c_mod=*/(short)0, c, /*reuse_a=*/false, /*reuse_b=*/false);
    }

    // C/D layout: VGPR j -> M = j + 8*laneHi, N = lane&15. Store the 8 valid cols.
    if (n < NG) {
        #pragma unroll
        for (int j = 0; j < 8; ++j) {
            const int m = j + 8 * laneHi;
            gates[(size_t)(rowBase + m) * NG + n] = c[j];
        }
    }
}

// ---------------------------------------------------------------------------
// Kernel 2: per-batch-row LSTM scans (hidden=1, serial), softmax over T,
// threshold/argmax selection. One block per b.
// ---------------------------------------------------------------------------
__device__ __forceinline__ float sigm_(float v) { return 1.0f / (1.0f + __expf(-v)); }
__device__ __forceinline__ float tanh_(float v) { return 2.0f / (1.0f + __expf(-2.0f * v)) - 1.0f; }

__global__ __launch_bounds__(256)
void scan_softmax_select(const float* __restrict__ gates,   // [B*T, NG]
                         const unsigned char* __restrict__ mask, // [B, T] bool
                         const float* __restrict__ Whh_f,   // [4]
                         const float* __restrict__ bih_f,
                         const float* __restrict__ bhh_f,
                         const float* __restrict__ Whh_b,
                         const float* __restrict__ bih_b,
                         const float* __restrict__ bhh_b,
                         float* __restrict__ att_out,       // [B, T]
                         float* __restrict__ w_out)         // [B, T] eff weights
{
    __shared__ float sf[T_];
    __shared__ float sb[T_];
    __shared__ float redv[256];
    __shared__ int   redi[256];

    const int b   = blockIdx.x;
    const int tid = threadIdx.x;
    const float* gb = gates + (size_t)b * T_ * NG;

    if (tid == 0) {                       // forward chain (wave 0)
        const float bi0 = bih_f[0] + bhh_f[0], bi1 = bih_f[1] + bhh_f[1];
        const float bi2 = bih_f[2] + bhh_f[2], bi3 = bih_f[3] + bhh_f[3];
        const float w0 = Whh_f[0], w1 = Whh_f[1], w2 = Whh_f[2], w3 = Whh_f[3];
        float h = 0.0f, cc = 0.0f;
        for (int t = 0; t < T_; ++t) {
            const float* gp = gb + t * NG;
            const float i = sigm_(gp[0] + bi0 + h * w0);
            const float f = sigm_(gp[1] + bi1 + h * w1);
            const float g = tanh_(gp[2] + bi2 + h * w2);
            const float o = sigm_(gp[3] + bi3 + h * w3);
            cc = f * cc + i * g;
            h  = o * tanh_(cc);
            sf[t] = h;
        }
    }
    if (tid == 64) {                      // backward chain (wave 2, runs concurrently)
        const float bi0 = bih_b[0] + bhh_b[0], bi1 = bih_b[1] + bhh_b[1];
        const float bi2 = bih_b[2] + bhh_b[2], bi3 = bih_b[3] + bhh_b[3];
        const float w0 = Whh_b[0], w1 = Whh_b[1], w2 = Whh_b[2], w3 = Whh_b[3];
        float h = 0.0f, cc = 0.0f;
        for (int t = T_ - 1; t >= 0; --t) {
            const float* gp = gb + t * NG + 4;
            const float i = sigm_(gp[0] + bi0 + h * w0);
            const float f = sigm_(gp[1] + bi1 + h * w1);
            const float g = tanh_(gp[2] + bi2 + h * w2);
            const float o = sigm_(gp[3] + bi3 + h * w3);
            cc = f * cc + i * g;
            h  = o * tanh_(cc);
            sb[t] = h;
        }
    }
    __syncthreads();

    // ----- softmax over T=512; each thread owns t0=tid, t1=tid+256 -----
    const int t0 = tid, t1 = tid + 256;
    const unsigned char* mb = mask + (size_t)b * T_;
    const float NEG_INF = -__builtin_inff();
    const float s0 = mb[t0] ? (sf[t0] + sb[t0]) : NEG_INF;
    const float s1 = mb[t1] ? (sf[t1] + sb[t1]) : NEG_INF;

    redv[tid] = fmaxf(s0, s1);
    __syncthreads();
    for (int off = 128; off > 0; off >>= 1) {
        if (tid < off) redv[tid] = fmaxf(redv[tid], redv[tid + off]);
        __syncthreads();
    }
    const float mx = redv[0];
    __syncthreads();

    const float e0 = __expf(s0 - mx), e1 = __expf(s1 - mx);
    redv[tid] = e0 + e1;
    __syncthreads();
    for (int off = 128; off > 0; off >>= 1) {
        if (tid < off) redv[tid] += redv[tid + off];
        __syncthreads();
    }
    const float inv = 1.0f / redv[0];
    __syncthreads();

    const float a0 = e0 * inv, a1 = e1 * inv;
    att_out[(size_t)b * T_ + t0] = a0;
    att_out[(size_t)b * T_ + t1] = a1;

    // ----- any(att >= THRESH) -----
    redi[tid] = (a0 >= THRESH_ ? 1 : 0) + (a1 >= THRESH_ ? 1 : 0);
    __syncthreads();
    for (int off = 128; off > 0; off >>= 1) {
        if (tid < off) redi[tid] += redi[tid + off];
        __syncthreads();
    }
    const int anySel = (redi[0] > 0);
    __syncthreads();

    // ----- argmax(att) with first-occurrence tie break -----
    float bv; int bi;
    if (a1 > a0) { bv = a1; bi = t1; } else { bv = a0; bi = t0; }
    redv[tid] = bv; redi[tid] = bi;
    __syncthreads();
    for (int off = 128; off > 0; off >>= 1) {
        if (tid < off) {
            const float vr = redv[tid + off]; const int ir = redi[tid + off];
            if (vr > redv[tid] || (vr == redv[tid] && ir < redi[tid])) {
                redv[tid] = vr; redi[tid] = ir;
            }
        }
        __syncthreads();
    }
    const int amax = redi[0];

    const float w0s = a0 * ((anySel ? (a0 >= THRESH_) : (t0 == amax)) ? 1.0f : 0.0f);
    const float w1s = a1 * ((anySel ? (a1 >= THRESH_) : (t1 == amax)) ? 1.0f : 0.0f);
    w_out[(size_t)b * T_ + t0] = w0s;
    w_out[(size_t)b * T_ + t1] = w1s;
}

// ---------------------------------------------------------------------------
// Kernel 3: out[b,f] = sum_t x[b,t,f] * w[b,t].  Second full read of x.
// grid (F/256, B), 256 threads; weights staged in LDS.
// ---------------------------------------------------------------------------
__global__ __launch_bounds__(256)
void weighted_sum(const float* __restrict__ x,
                  const float* __restrict__ w,      // [B, T]
                  float* __restrict__ out)          // [B, F]
{
    __shared__ float ws_s[T_];
    const int b = blockIdx.y;
    const int f = blockIdx.x * 256 + threadIdx.x;
    for (int t = threadIdx.x; t < T_; t += 256) ws_s[t] = w[(size_t)b * T_ + t];
    __syncthreads();

    const float* xb = x + (size_t)b * T_ * F_ + f;
    float acc = 0.0f;
    #pragma unroll 4
    for (int t = 0; t < T_; ++t) acc = fmaf(xb[(size_t)t * F_], ws_s[t], acc);
    out[(size_t)b * F_ + f] = acc;
}

// ---------------------------------------------------------------------------
extern "C" void kernel_launch(void* const* d_in, const int* in_sizes, int n_in,
                              void* d_out, int out_size, void* d_ws, size_t ws_size,
                              hipStream_t stream) {
    const float*         x     = (const float*)d_in[0];
    const unsigned char* mask  = (const unsigned char*)d_in[1];
    const float*         Wih_f = (const float*)d_in[2];
    const float*         Whh_f = (const float*)d_in[3];
    const float*         bih_f = (const float*)d_in[4];
    const float*         bhh_f = (const float*)d_in[5];
    const float*         Wih_b = (const float*)d_in[6];
    const float*         Whh_b = (const float*)d_in[7];
    const float*         bih_b = (const float*)d_in[8];
    const float*         bhh_b = (const float*)d_in[9];

    float* out = (float*)d_out;          // [B, F] = 49152 floats
    float* att = out + B_ * F_;          // [B, T] = 32768 floats

    float* gates = (float*)d_ws;                     // [B*T, NG] = 1 MB
    float* wsel  = gates + (size_t)B_ * T_ * NG;     // [B, T]    = 128 KB

    gates_gemm_wmma<<<(B_ * T_) / 128, 256, 0, stream>>>(x, Wih_f, Wih_b, gates);
    scan_softmax_select<<<B_, 256, 0, stream>>>(gates, mask,
                                                Whh_f, bih_f, bhh_f,
                                                Whh_b, bih_b, bhh_b,
                                                att, wsel);
    weighted_sum<<<dim3(F_ / 256, B_), 256, 0, stream>>>(x, wsel, out);
}